// SparseVolumeReconstructionLinear_3496103379278
// MI455X (gfx1250) — compile-verified
//
#include <hip/hip_runtime.h>

typedef float v2f __attribute__((ext_vector_type(2)));
typedef float v8f __attribute__((ext_vector_type(8)));

#define SIZE_   161
#define MARGIN_ 3
#define BZM     (SIZE_ + 2 * MARGIN_)        // 167
#define BZ2     (BZM / 2)                    // 83
#define GX      (BZM / 2 + 1)                // 84
#define P_TOT   (SIZE_ * (SIZE_ / 2 + 1))    // 13041
#define BATCH_  256

__global__ __launch_bounds__(256) void svr_kernel(
    const float* __restrict__ inp,     // (256, 8)
    const float* __restrict__ weight,  // (W, 8, 2)
    const float* __restrict__ bias,    // (W, 2)
    const int*   __restrict__ g3d32,   // (167,167,84) int64 viewed as int32 pairs
    const float* __restrict__ rot,     // (256, 3, 3)
    const float* __restrict__ coord,   // (P, 2)
    const int*   __restrict__ max_r_p,
    float*       __restrict__ out)     // (256, P, 2)
{
    const int b    = blockIdx.y;                         // uniform per block
    const int gid  = blockIdx.x * blockDim.x + threadIdx.x;
    const int p    = (gid < P_TOT) ? gid : (P_TOT - 1);  // clamp: keep all lanes live for WMMA
    const int lane = threadIdx.x & 31;
    const bool hi  = lane >= 16;
    const int nlane = lane & 15;

    // ---- block-uniform state (scalar loads) ----
    const float* R = rot + b * 9;
    const float r00 = R[0], r01 = R[1];
    const float r10 = R[3], r11 = R[4];
    const float r20 = R[6], r21 = R[7];
    float in[8];
    #pragma unroll
    for (int d = 0; d < 8; ++d) in[d] = inp[b * 8 + d];
    int mr = *max_r_p;
    if (mr > (BZM - 2 * MARGIN_) / 2) mr = (BZM - 2 * MARGIN_) / 2;
    const float mr2 = (float)(mr * mr);

    // ---- per-point geometry ----
    const float2 xy = ((const float2*)coord)[p];
    const bool valid = (xy.x * xy.x + xy.y * xy.y) <= mr2;
    float cx = r00 * xy.x + r01 * xy.y;
    float cy = r10 * xy.x + r11 * xy.y;
    float cz = r20 * xy.x + r21 * xy.y;
    const bool flip = cx < 0.0f;
    if (flip) { cx = -cx; cy = -cy; cz = -cz; }
    if (!valid) { cx = 0.0f; cy = 0.0f; cz = 0.0f; }

    const float fx = floorf(cx), fy = floorf(cy), fz = floorf(cz);
    const float tx = cx - fx, ty = cy - fy, tz = cz - fz;
    const int ix = (int)fx;
    const int iy = (int)fy + BZ2;
    const int iz = (int)fz + BZ2;

    const float wx[2] = {1.0f - tx, tx};
    const float wy[2] = {1.0f - ty, ty};
    const float wz[2] = {1.0f - tz, tz};

    // ---- gather-heavy corner loop: accumulate trilinear-weighted weight rows ----
    float sw[2][8];   // [channel][d]
    float sb[2] = {0.0f, 0.0f};
    #pragma unroll
    for (int d = 0; d < 8; ++d) { sw[0][d] = 0.0f; sw[1][d] = 0.0f; }

    #pragma unroll
    for (int dz = 0; dz < 2; ++dz) {
        #pragma unroll
        for (int dy = 0; dy < 2; ++dy) {
            #pragma unroll
            for (int dx = 0; dx < 2; ++dx) {
                const int lin = ((iz + dz) * BZM + (iy + dy)) * GX + (ix + dx);
                const int j = g3d32[2 * lin];  // low dword of little-endian int64
                const float w = wx[dx] * wy[dy] * wz[dz];
                if (j >= 0) {
                    const float4* wr = (const float4*)(weight + (size_t)j * 16);
                    const float4 q0 = wr[0], q1 = wr[1], q2 = wr[2], q3 = wr[3];
                    sw[0][0] += w * q0.x; sw[1][0] += w * q0.y;
                    sw[0][1] += w * q0.z; sw[1][1] += w * q0.w;
                    sw[0][2] += w * q1.x; sw[1][2] += w * q1.y;
                    sw[0][3] += w * q1.z; sw[1][3] += w * q1.w;
                    sw[0][4] += w * q2.x; sw[1][4] += w * q2.y;
                    sw[0][5] += w * q2.z; sw[1][5] += w * q2.w;
                    sw[0][6] += w * q3.x; sw[1][6] += w * q3.y;
                    sw[0][7] += w * q3.z; sw[1][7] += w * q3.w;
                    const float2 bb = ((const float2*)bias)[j];
                    sb[0] += w * bb.x; sb[1] += w * bb.y;
                }
            }
        }
    }

    // ---- WMMA projection: out_c(p) = sum_d in[d]*sw[c][d] + sb[c], fp32, K=8 via 2 chained 16x16x4 ----
    // A[m,k] = in[k] for all m (16x4 f32 A layout: lanes 0-15 hold k={0,1}, lanes 16-31 k={2,3})
    v2f aLo, aHi;
    aLo.x = hi ? in[2] : in[0];
    aLo.y = hi ? in[3] : in[1];
    aHi.x = hi ? in[6] : in[4];
    aHi.y = hi ? in[7] : in[5];

    float vals[2];
    #pragma unroll
    for (int c = 0; c < 2; ++c) {
        float vt[2];
        #pragma unroll
        for (int t = 0; t < 2; ++t) {        // tile t covers wave points 16t .. 16t+15
            const int src = t * 16 + nlane;  // source lane owning column's point
            // B[k,n] = sw[c][k] of point n (4x16 f32 B layout, transpose-symmetric to A)
            v2f bLo, bHi;
            {
                const float g0 = __shfl(sw[c][0], src, 32);
                const float g2 = __shfl(sw[c][2], src, 32);
                bLo.x = hi ? g2 : g0;
                const float g1 = __shfl(sw[c][1], src, 32);
                const float g3 = __shfl(sw[c][3], src, 32);
                bLo.y = hi ? g3 : g1;
                const float g4 = __shfl(sw[c][4], src, 32);
                const float g6 = __shfl(sw[c][6], src, 32);
                bHi.x = hi ? g6 : g4;
                const float g5 = __shfl(sw[c][5], src, 32);
                const float g7 = __shfl(sw[c][7], src, 32);
                bHi.y = hi ? g7 : g5;
            }
            const float cc = __shfl(sb[c], src, 32);  // C[m,n] = sb of point n (row-invariant)
            v8f acc;
            #pragma unroll
            for (int r = 0; r < 8; ++r) acc[r] = cc;
            acc = __builtin_amdgcn_wmma_f32_16x16x4_f32(false, aLo, false, bLo,
                                                        (short)0, acc, false, false);
            acc = __builtin_amdgcn_wmma_f32_16x16x4_f32(false, aHi, false, bHi,
                                                        (short)0, acc, false, false);
            vt[t] = acc[0];  // A rows identical -> every row of D equals out_c(col); col = lane%16
        }
        vals[c] = hi ? vt[1] : vt[0];
    }

    // ---- epilogue ----
    float re = vals[0];
    float im = flip ? -vals[1] : vals[1];
    if (!valid) { re = 0.0f; im = 0.0f; }
    if (gid < P_TOT) {
        float2 o; o.x = re; o.y = im;
        ((float2*)out)[(size_t)b * P_TOT + gid] = o;
    }
}

extern "C" void kernel_launch(void* const* d_in, const int* in_sizes, int n_in,
                              void* d_out, int out_size, void* d_ws, size_t ws_size,
                              hipStream_t stream) {
    (void)in_sizes; (void)n_in; (void)out_size; (void)d_ws; (void)ws_size;
    const float* inp    = (const float*)d_in[0];
    const float* weight = (const float*)d_in[1];
    const float* bias   = (const float*)d_in[2];
    const int*   g3d32  = (const int*)d_in[3];   // int64 grid, read low dwords
    const float* rot    = (const float*)d_in[4];
    const float* coord  = (const float*)d_in[5];
    const int*   max_r  = (const int*)d_in[6];
    float*       out    = (float*)d_out;

    dim3 block(256, 1, 1);
    dim3 grid((P_TOT + 255) / 256, BATCH_, 1);
    svr_kernel<<<grid, block, 0, stream>>>(inp, weight, bias, g3d32, rot, coord, max_r, out);
}